// Memory_Pooling_Layer_19877108646275
// MI455X (gfx1250) — compile-verified
//
#include <hip/hip_runtime.h>

// ---------------------------------------------------------------------------
// Memory_Pooling_Layer for MI455X (gfx1250, wave32)
//
// Key observation: the reference applies softmax over a singleton axis, so the
// pooling matrix C is exactly all-ones. Outputs reduce to:
//   new_node_set[b,k,:] = leakyrelu((sum_n node_set[b,n,:]) @ lin_w + lin_b)
//   new_adj[b,k,j]      = sum_{n,m} adj[b,n,m]
// Dominant cost: streaming adj (536 MB) once -> ~23 us at 23.3 TB/s. adj is
// read-once, so its loads are non-temporal. The tiny 16x128x128 linear layer
// uses V_WMMA_F32_16X16X4_F32 with a zero-padded 16-row A matrix (no EXEC
// divergence inside the WMMA loop).
// ---------------------------------------------------------------------------

typedef __attribute__((ext_vector_type(2))) float v2f;
typedef __attribute__((ext_vector_type(4))) float v4f;
typedef __attribute__((ext_vector_type(8))) float v8f;

#define B_DIM 8
#define N_IN 4096
#define D_IN 128
#define N_OUT 256
#define D_OUT 128

// ---- adj total sum, stage 1: one partial per block -------------------------
// 8192 blocks * 256 threads. Block i sums a contiguous 16384-float slice
// (64 KB) of adj via non-temporal 128-bit loads. adj has 8*4096*4096 = 2^27
// floats, 2^27 / 8192 = 16384 per block. Deterministic LDS tree reduce.
__global__ __launch_bounds__(256) void adj_partial_kernel(
    const v4f* __restrict__ adj4, float* __restrict__ part) {
  const int t = threadIdx.x;
  const long long base4 = (long long)blockIdx.x * 4096;  // float4 units
  float sum = 0.f;
#pragma unroll
  for (int i = 0; i < 16; ++i) {
    v4f v = __builtin_nontemporal_load(&adj4[base4 + t + i * 256]);
    sum += (v.x + v.y) + (v.z + v.w);
  }
  __shared__ float red[256];
  red[t] = sum;
  __syncthreads();
  for (int s = 128; s > 0; s >>= 1) {
    if (t < s) red[t] += red[t + s];
    __syncthreads();
  }
  if (t == 0) part[blockIdx.x] = red[0];
}

// ---- adj total sum, stage 2: fold 1024 partials per batch ------------------
__global__ __launch_bounds__(256) void adj_final_kernel(
    const float* __restrict__ part, float* __restrict__ adjSum) {
  const int b = blockIdx.x;
  const int t = threadIdx.x;
  float sum = part[b * 1024 + t] + part[b * 1024 + t + 256] +
              part[b * 1024 + t + 512] + part[b * 1024 + t + 768];
  __shared__ float red[256];
  red[t] = sum;
  __syncthreads();
  for (int s = 128; s > 0; s >>= 1) {
    if (t < s) red[t] += red[t + s];
    __syncthreads();
  }
  if (t == 0) adjSum[b] = red[0];
}

// ---- node_set column sums, stage 1 ----------------------------------------
// 128 blocks (b*16 + chunk) * 128 threads. Each block sums 256 rows of 128
// columns; thread d accumulates column d (fully coalesced row reads).
__global__ __launch_bounds__(128) void node_partial_kernel(
    const float* __restrict__ ns, float* __restrict__ part) {
  const int blk = blockIdx.x;  // b*16 + chunk
  const int d = threadIdx.x;
  const long long base = (long long)blk * 256 * D_IN;
  float sum = 0.f;
  for (int r = 0; r < 256; ++r) sum += ns[base + r * D_IN + d];
  part[blk * D_IN + d] = sum;
}

// ---- node_set column sums, stage 2 ----------------------------------------
// 16 blocks: blocks 0..7 fold the 16 chunk-partials of batch b; blocks 8..15
// write zeros so sVec is a full 16x128 zero-padded WMMA A-matrix (the
// workspace is poisoned with 0xAA, so explicit zeroing is mandatory).
__global__ __launch_bounds__(128) void node_final_kernel(
    const float* __restrict__ part, float* __restrict__ sVec) {
  const int b = blockIdx.x;
  const int d = threadIdx.x;
  float sum = 0.f;
  if (b < B_DIM) {
#pragma unroll
    for (int c = 0; c < 16; ++c) sum += part[(b * 16 + c) * D_IN + d];
  }
  sVec[b * D_IN + d] = sum;
}

// ---- linear layer via V_WMMA_F32_16X16X4_F32 ------------------------------
// A = sVec (16x128, rows 8..15 zero), B = lin_w (128x128), D -> y rows 0..7.
// One block, 8 waves; wave w owns output columns [16w, 16w+16).
// A layout (32-bit A 16x4): lanes 0-15 hold K={k,k+1}, lanes 16-31 K={k+2,k+3}
// for row M = lane&15. B mirrored (row-striped across lanes per VGPR).
// D layout: VGPR r = row M=r for lanes 0-15 (M=8+r for lanes 16-31).
// All loads are unconditional -> no EXEC churn between WMMAs.
__global__ __launch_bounds__(256) void lin_wmma_kernel(
    const float* __restrict__ sVec, const float* __restrict__ lin_w,
    const float* __restrict__ lin_b, float* __restrict__ yVec) {
  const int lane = threadIdx.x & 31;
  const int wave = threadIdx.x >> 5;
  const int half = lane >> 4;
  const int m = lane & 15;
  const int n = wave * 16 + m;  // output column 0..127

  const float* aRow = sVec + m * D_IN + 2 * half;   // A[m][2*half + k]
  const float* bCol = lin_w + 2 * half * D_OUT + n; // B[2*half + k][n]

  v8f acc = {};
#pragma unroll 4
  for (int k = 0; k < D_IN; k += 4) {
    v2f a, bmat;
    a.x = aRow[k];
    a.y = aRow[k + 1];
    bmat.x = bCol[k * D_OUT];
    bmat.y = bCol[(k + 1) * D_OUT];
    // 8 args: (neg_a, A, neg_b, B, c_mod, C, reuse_a, reuse_b)
    acc = __builtin_amdgcn_wmma_f32_16x16x4_f32(
        false, a, false, bmat, (short)0, acc, false, false);
  }

  if (lane < 16) {
    const float bias = lin_b[n];
#pragma unroll
    for (int r = 0; r < B_DIM; ++r) {
      float v = acc[r] + bias;
      v = (v >= 0.f) ? v : 0.01f * v;
      yVec[r * D_OUT + n] = v;
    }
  }
}

// ---- broadcast both outputs ------------------------------------------------
// out[0 .. 262143]       : new_node_set[b,k,dout] = y[b,dout]
// out[262144 .. 786431]  : new_adj[b,k,j]         = adjSum[b]
__global__ __launch_bounds__(256) void broadcast_kernel(
    const float* __restrict__ yVec, const float* __restrict__ adjSum,
    float* __restrict__ out) {
  const int idx = blockIdx.x * blockDim.x + threadIdx.x;
  const int NODE_TOTAL = B_DIM * N_OUT * D_OUT;           // 262144
  const int TOTAL = NODE_TOTAL + B_DIM * N_OUT * N_OUT;   // 786432
  if (idx >= TOTAL) return;
  if (idx < NODE_TOTAL) {
    const int b = idx >> 15;       // / (256*128)
    const int dout = idx & 127;
    out[idx] = yVec[b * D_OUT + dout];
  } else {
    const int j = idx - NODE_TOTAL;
    const int b = j >> 16;         // / (256*256)
    out[idx] = adjSum[b];
  }
}

extern "C" void kernel_launch(void* const* d_in, const int* in_sizes, int n_in,
                              void* d_out, int out_size, void* d_ws,
                              size_t ws_size, hipStream_t stream) {
  const float* node_set = (const float*)d_in[0];
  const float* adj = (const float*)d_in[1];
  // d_in[2..4] (centroids, conv_w, conv_b) are dead: softmax over a singleton
  // axis makes C identically 1.0, so they never affect the output.
  const float* lin_w = (const float*)d_in[5];
  const float* lin_b = (const float*)d_in[6];
  float* out = (float*)d_out;

  float* ws = (float*)d_ws;
  float* adjPart = ws;              // 8192 floats
  float* adjSum = ws + 8192;        // 8
  float* nodePart = ws + 8200;      // 16384
  float* sVec = ws + 24584;         // 2048 (16x128, rows 8..15 zeroed)
  float* yVec = ws + 26632;         // 1024  (total ~110 KB)

  adj_partial_kernel<<<8192, 256, 0, stream>>>((const v4f*)adj, adjPart);
  adj_final_kernel<<<B_DIM, 256, 0, stream>>>(adjPart, adjSum);
  node_partial_kernel<<<B_DIM * 16, 128, 0, stream>>>(node_set, nodePart);
  node_final_kernel<<<16, 128, 0, stream>>>(nodePart, sVec);
  lin_wmma_kernel<<<1, 256, 0, stream>>>(sVec, lin_w, lin_b, yVec);
  broadcast_kernel<<<3072, 256, 0, stream>>>(yVec, adjSum, out);
}